// TurboQuantValueMSE_2250562863576
// MI455X (gfx1250) — compile-verified
//
#include <hip/hip_runtime.h>
#include <stdint.h>

// ============================================================================
// TurboQuant 4-bit quantize->dequantize (group=32, head_dim=128) for gfx1250.
//
// Streaming kernel: 512MB f32 in + 512MB f32 out (fp16 values widened)
// ~= 1GB @ 23.3 TB/s => ~44us floor. No matrix math; optimize data movement:
//   - CDNA5 async global->LDS (GLOBAL_LOAD_ASYNC_TO_LDS_B128, ASYNCcnt)
//     double-buffered per wave (1KB tiles, 16KB LDS/block out of 320KB/WGP)
//   - non-temporal b128 stores (1GB stream >> 192MB L2)
//   - group(32) min/max via two ds-swizzle xor hops across 4 lanes
// Exactness vs reference:
//   - IEEE f32 division for (x-min)/scale and (max-min)/15
//   - rintf == jnp.round (round half to even)
//   - fp16 dequant as separate mul then add (fp contract off), RTE
// ============================================================================

typedef __attribute__((ext_vector_type(4))) float f32x4;
typedef int i32x4v __attribute__((vector_size(16)));   // matches builtin param type

constexpr int THREADS    = 256;            // 8 waves (wave32)
constexpr int EPL        = 8;              // elements per lane
constexpr int WAVE_ELEMS = 32 * EPL;       // 256 floats per wave-tile (1KB)
constexpr int BLOCKS     = 2048;

#if defined(__AMDGCN__) && __has_builtin(__builtin_amdgcn_global_load_async_to_lds_b128)
#define USE_ASYNC 1
#else
#define USE_ASYNC 0
#endif

#if USE_ASYNC
// Builtin signature (from hipcc diagnostic): first param is
//   int __attribute__((vector_size(16))) __device__ *   (v4i in addrspace(1))
// second is the LDS-side pointer (addrspace(3)); then (int, int) immediates.
// We integer-launder the pointers: flat addresses of __shared__ truncate to the
// LDS offset per CDNA5 aperture rules, so the low 32 bits are the DS address.
#define GLB1(p)  ((__attribute__((address_space(1))) i32x4v*)(unsigned long long)(uintptr_t)(p))
#define LDS3(o)  ((__attribute__((address_space(3))) i32x4v*)(unsigned)(o))
#if __has_builtin(__builtin_amdgcn_s_wait_asynccnt)
#define WAIT_ASYNC(n) do { __builtin_amdgcn_s_wait_asynccnt(n); asm volatile("" ::: "memory"); } while (0)
#else
#define WAIT_ASYNC(n) asm volatile("s_wait_asynccnt %0" :: "n"(n) : "memory")
#endif
#endif

__device__ __forceinline__ void qdq_group8(const float v[EPL], float* __restrict__ outp) {
#pragma clang fp contract(off)  // fp16 mul and add must round separately (matches ref)
  float mn = v[0], mx = v[0];
#pragma unroll
  for (int i = 1; i < EPL; ++i) { mn = fminf(mn, v[i]); mx = fmaxf(mx, v[i]); }
  // A 32-element group spans 4 adjacent lanes: xor-shuffle reduction.
  mn = fminf(mn, __shfl_xor(mn, 1));
  mn = fminf(mn, __shfl_xor(mn, 2));
  mx = fmaxf(mx, __shfl_xor(mx, 1));
  mx = fmaxf(mx, __shfl_xor(mx, 2));
  const float scale = fmaxf((mx - mn) / 15.0f, 1e-8f);   // IEEE div, matches ref
  const _Float16 mh = (_Float16)mn;     // RTE f32->f16, matches astype(f16)
  const _Float16 sh = (_Float16)scale;
  float o[EPL];
#pragma unroll
  for (int i = 0; i < EPL; ++i) {
    float r = rintf((v[i] - mn) / scale);   // round half to even == jnp.round
    r = fminf(fmaxf(r, 0.0f), 15.0f);       // clip to [0,15] (v_med3)
    const _Float16 q = (_Float16)r;         // exact small integer
    const _Float16 p = q * sh;              // fp16 RTE multiply
    const _Float16 d = p + mh;              // fp16 RTE add
    o[i] = (float)d;
  }
  f32x4 lo = {o[0], o[1], o[2], o[3]};
  f32x4 hi = {o[4], o[5], o[6], o[7]};
  f32x4* ov = (f32x4*)outp;                 // 32B-aligned (lane base is 8-float aligned)
  __builtin_nontemporal_store(lo, ov);
  __builtin_nontemporal_store(hi, ov + 1);
}

__global__ void __launch_bounds__(THREADS)
turboquant_qdq_kernel(const float* __restrict__ x, float* __restrict__ out,
                      long long nTiles) {
  const int lane = threadIdx.x & 31;
  const int wv   = threadIdx.x >> 5;
  const long long wstride = (long long)gridDim.x * (THREADS / 32);
  const long long gw = (long long)blockIdx.x * (THREADS / 32) + wv;

#if USE_ASYNC
  __shared__ float lbuf[THREADS * EPL * 2];          // 8 waves x 2 bufs x 1KB = 16KB
  float* wbase = &lbuf[wv * 2 * WAVE_ELEMS];
  const unsigned ldsLane =
      (unsigned)(uintptr_t)wbase + (unsigned)(lane * EPL * 4);

  long long t = gw;
  if (t < nTiles) {                                  // prologue: stage tile 0 -> buf 0
    const float* g = x + t * WAVE_ELEMS + (long long)lane * EPL;
    __builtin_amdgcn_global_load_async_to_lds_b128(GLB1(g),     LDS3(ldsLane),      0, 0);
    __builtin_amdgcn_global_load_async_to_lds_b128(GLB1(g + 4), LDS3(ldsLane + 16), 0, 0);
  }
  int k = 0;
  for (; t < nTiles; t += wstride, k ^= 1) {
    const long long tn = t + wstride;
    if (tn < nTiles) {                               // prefetch next tile -> other buf
      const float* g = x + tn * WAVE_ELEMS + (long long)lane * EPL;
      const unsigned lo = ldsLane + (unsigned)((k ^ 1) * WAVE_ELEMS * 4);
      __builtin_amdgcn_global_load_async_to_lds_b128(GLB1(g),     LDS3(lo),      0, 0);
      __builtin_amdgcn_global_load_async_to_lds_b128(GLB1(g + 4), LDS3(lo + 16), 0, 0);
      WAIT_ASYNC(2);   // async ops complete in order: ours (older 2) are done
    } else {
      WAIT_ASYNC(0);
    }
    const float* src = wbase + k * WAVE_ELEMS + lane * EPL;
    const f32x4 a = *(const f32x4*)src;              // ds_load_b128
    const f32x4 b = *(const f32x4*)(src + 4);
    float v[EPL];
#pragma unroll
    for (int i = 0; i < 4; ++i) { v[i] = a[i]; v[i + 4] = b[i]; }
    qdq_group8(v, out + t * WAVE_ELEMS + (long long)lane * EPL);
  }
#else
  // Fallback: direct non-temporal b128 loads (still bandwidth-optimal).
  for (long long t = gw; t < nTiles; t += wstride) {
    const f32x4* src = (const f32x4*)(x + t * WAVE_ELEMS + (long long)lane * EPL);
    const f32x4 a = __builtin_nontemporal_load(src);
    const f32x4 b = __builtin_nontemporal_load(src + 1);
    float v[EPL];
#pragma unroll
    for (int i = 0; i < 4; ++i) { v[i] = a[i]; v[i + 4] = b[i]; }
    qdq_group8(v, out + t * WAVE_ELEMS + (long long)lane * EPL);
  }
#endif
}

extern "C" void kernel_launch(void* const* d_in, const int* in_sizes, int n_in,
                              void* d_out, int out_size, void* d_ws, size_t ws_size,
                              hipStream_t stream) {
  (void)n_in; (void)out_size; (void)d_ws; (void)ws_size;
  const float* x = (const float*)d_in[0];
  float* out = (float*)d_out;
  const long long total = (long long)in_sizes[0];       // 4*32*8192*128 = 134,217,728
  const long long nTiles = total / WAVE_ELEMS;          // divisible: vec len 128, even vec count
  turboquant_qdq_kernel<<<dim3(BLOCKS), dim3(THREADS), 0, stream>>>(x, out, nTiles);
}